// NeuralGraphFingerprintModel_52682068853040
// MI455X (gfx1250) — compile-verified
//
#include <hip/hip_runtime.h>
#include <stdint.h>

// ---------------------------------------------------------------------------
// Problem constants (match reference)
// ---------------------------------------------------------------------------
constexpr int BATCH = 128;
constexpr int NATOM = 128;
constexpr int FA    = 64;
constexpr int FB    = 16;
constexpr int K1    = 2 * FA + FB;   // 144
constexpr int HID   = 512;
constexpr int NOUT  = 128;

// LDS row strides (floats), padded against bank conflicts
constexpr int XS   = 148;  // Xcat rows      (bank stride 20 -> 16 distinct)
constexpr int W1TS = 148;  // W1^T rows [64 x 144]   (contiguous K per column)
constexpr int HS   = 68;   // H tile rows
constexpr int W2TS = 68;   // W2^T rows [128 x 64]

// LDS layout (bytes). adj aliases the [W1T|H|W2T] region (disjoint in time).
constexpr size_t OFF_X   = 0;
constexpr size_t OFF_R   = (size_t)NATOM * XS * 4;                  // 75,776
constexpr size_t OFF_W1T = OFF_R;                                   // 37,888 B
constexpr size_t OFF_H   = OFF_R + (size_t)64 * W1TS * 4;           // +37,888
constexpr size_t OFF_W2T = OFF_H + (size_t)NATOM * HS * 4;          // +34,816
constexpr size_t OFF_DEG = OFF_W2T + (size_t)NOUT * W2TS * 4;       // 183,296
constexpr size_t OFF_RED = OFF_DEG + 512;
constexpr size_t LDS_BYTES = OFF_RED + 512;                         // 184,320

// ---------------------------------------------------------------------------
// Vector types
// ---------------------------------------------------------------------------
typedef float v2f  __attribute__((ext_vector_type(2)));
typedef float v8f  __attribute__((ext_vector_type(8)));
typedef unsigned int u32x4 __attribute__((ext_vector_type(4)));
typedef int   i32x8 __attribute__((ext_vector_type(8)));
typedef int   i32x4 __attribute__((ext_vector_type(4)));

__global__ __launch_bounds__(256) void ngf_fused(
    const float* __restrict__ atomf, const float* __restrict__ bondf,
    const float* __restrict__ adjf,  const float* __restrict__ W1,
    const float* __restrict__ b1,    const float* __restrict__ W2,
    const float* __restrict__ b2,    float* __restrict__ outp)
{
  extern __shared__ __align__(16) char smem[];
  float* s_x   = (float*)(smem + OFF_X);
  float* s_adj = (float*)(smem + OFF_R);
  float* s_w1t = (float*)(smem + OFF_W1T);   // [64][W1TS]  (col-major W1 chunk)
  float* s_h   = (float*)(smem + OFF_H);     // [128][HS]
  float* s_w2t = (float*)(smem + OFF_W2T);   // [128][W2TS] (col-major W2 chunk)
  float* s_deg = (float*)(smem + OFF_DEG);
  float* s_red = (float*)(smem + OFF_RED);

  const int tid  = threadIdx.x;
  const int b    = blockIdx.x;
  const int lane = tid & 31;
  const int wv   = tid >> 5;          // 8 waves
  const int ra   = lane & 15;         // row-in-tile (A) / col (B,C,D)
  const int kg   = lane >> 4;         // K half-group (0: K0,K1 | 1: K2,K3)
  const int m0   = wv << 4;           // this wave's 16-row strip

  const v8f vzero = {0.f,0.f,0.f,0.f,0.f,0.f,0.f,0.f};

  // ---- Stage 0: TDM async tensor load  adj[b] (128x128 f32) -> LDS --------
  if (wv == 0) {
    uint64_t ga = (uint64_t)(uintptr_t)(adjf + (size_t)b * NATOM * NATOM);
    uint32_t lo = (uint32_t)(uintptr_t)s_adj;   // flat-shared low 32b = LDS off
    // D# group0: count=1 | lds_addr | global_addr[56:0] | type=2
    u32x4 g0 = { 1u, lo, (uint32_t)ga,
                 (uint32_t)((ga >> 32) & 0x01FFFFFFu) | (2u << 30) };
    // D# group1: data_size=4B; tensor_dim0=128, tensor_dim1=128,
    //            tile_dim0=128, tile_dim1=128, tensor_dim0_stride=128
    i32x8 g1 = { 0x00020000, (int)(128u << 16), (int)(128u << 16),
                 (int)(128u << 16), 128, 128, 0, 0 };
    i32x4 gz4 = { 0, 0, 0, 0 };
    i32x8 gz8 = { 0, 0, 0, 0, 0, 0, 0, 0 };
    __builtin_amdgcn_tensor_load_to_lds(g0, g1, gz4, gz4, gz8, 0);
    __builtin_amdgcn_s_wait_tensorcnt(0);
  }
  if (tid < NOUT) s_red[tid] = 0.0f;
  __syncthreads();

  // ---- Stage A1: deg[j] = sum_i adj[b,i,j] --------------------------------
  if (tid < NATOM) {
    float s = 0.f;
    #pragma unroll 8
    for (int i = 0; i < NATOM; ++i) s += s_adj[i * NATOM + tid];
    s_deg[tid] = s;
  }
  __syncthreads();

  // ---- Stage A2: Xcat[:,0:64]=atom, [:,64:128]=atom*deg -------------------
  const float* atomB = atomf + (size_t)b * NATOM * FA;
  for (int idx = tid; idx < NATOM * FA; idx += 256) {
    int j = idx >> 6, f = idx & 63;
    float v = atomB[idx];
    s_x[j * XS + f]      = v;
    s_x[j * XS + FA + f] = v * s_deg[j];
  }

  // ---- Stage A3: Xcat[:,128:144] = sum_i adj[b,i,j]*bond[b,i,j,:] ---------
  const float4* bondB =
      (const float4*)(bondf + (size_t)b * NATOM * NATOM * FB);
  for (int u = tid; u < NATOM * 4; u += 256) {       // (j, f-quad)
    int j = u >> 2, fq = u & 3;
    float4 acc = {0.f, 0.f, 0.f, 0.f};
    for (int i = 0; i < NATOM; ++i) {
      float a = s_adj[i * NATOM + j];
      if (a != 0.f) {                                // ~15% dense: skip loads
        float4 v = bondB[(i * NATOM + j) * 4 + fq];
        acc.x += a * v.x; acc.y += a * v.y;
        acc.z += a * v.z; acc.w += a * v.w;
      }
    }
    *(float4*)(s_x + j * XS + 2 * FA + 4 * fq) = acc;
  }

  // ---- Stage B: fused GEMM1 (relu) + GEMM2, hidden chunked by 64 ----------
  v8f acc2[8];                       // persistent 16x16 tiles of H@W2 rows
  #pragma unroll
  for (int i = 0; i < 8; ++i) acc2[i] = vzero;

  #pragma unroll 1
  for (int c = 0; c < HID / 64; ++c) {
    __syncthreads();                 // prev chunk done; adj region retired
    // W1 chunk, stored transposed: s_w1t[n][k] = W1[k][64c+n]
    const float* W1c = W1 + c * 64;                   // [144 x 512] cols
    for (int idx = tid; idx < K1 * 64; idx += 256) {
      int k = idx >> 6, n = idx & 63;                 // global-coalesced read
      s_w1t[n * W1TS + k] = W1c[k * HID + n];
    }
    // W2 chunk, stored transposed: s_w2t[n][k] = W2[64c+k][n]
    const float* W2c = W2 + (size_t)c * 64 * NOUT;    // [512 x 128] rows
    for (int idx = tid; idx < 64 * NOUT; idx += 256) {
      int k = idx >> 7, n = idx & 127;                // global-coalesced read
      s_w2t[n * W2TS + k] = W2c[idx];
    }
    __syncthreads();

    // GEMM1: Htile[m0:m0+16, 0:64] = relu(Xcat @ W1c + b1c)
    v8f acc1[4];
    #pragma unroll
    for (int i = 0; i < 4; ++i) acc1[i] = vzero;
    #pragma unroll 6
    for (int t = 0; t < K1 / 4; ++t) {
      int k = 4 * t;
      v2f a = *(const v2f*)(s_x + (m0 + ra) * XS + k + 2 * kg);
      #pragma unroll
      for (int ni = 0; ni < 4; ++ni) {
        v2f bb = *(const v2f*)(s_w1t + (ni * 16 + ra) * W1TS + k + 2 * kg);
        acc1[ni] = __builtin_amdgcn_wmma_f32_16x16x4_f32(
            false, a, false, bb, (short)0, acc1[ni], false, false);
      }
    }
    // bias + relu, D-layout (VGPR r -> row r / r+8) into this wave's H rows
    #pragma unroll
    for (int ni = 0; ni < 4; ++ni) {
      float bias = b1[c * 64 + ni * 16 + ra];
      #pragma unroll
      for (int r = 0; r < 8; ++r) {
        int row = m0 + (kg << 3) + r;
        float v = acc1[ni][r] + bias;
        s_h[row * HS + ni * 16 + ra] = fmaxf(v, 0.f);
      }
    }
    // GEMM2: acc2 += Htile[m0:m0+16, 0:64] @ W2c   (same-wave H rows only)
    #pragma unroll 4
    for (int t = 0; t < 16; ++t) {
      int k = 4 * t;
      v2f a = *(const v2f*)(s_h + (m0 + ra) * HS + k + 2 * kg);
      #pragma unroll
      for (int ni = 0; ni < 8; ++ni) {
        v2f bb = *(const v2f*)(s_w2t + (ni * 16 + ra) * W2TS + k + 2 * kg);
        acc2[ni] = __builtin_amdgcn_wmma_f32_16x16x4_f32(
            false, a, false, bb, (short)0, acc2[ni], false, false);
      }
    }
  }

  // ---- Stage C: out[b,:] = sum_rows(H@W2) + N*b2 --------------------------
  #pragma unroll
  for (int ni = 0; ni < 8; ++ni) {
    float s = 0.f;
    #pragma unroll
    for (int r = 0; r < 8; ++r) s += acc2[ni][r];   // 8 rows in this half
    s += __shfl_xor(s, 16, 32);                     // join the two row halves
    if (lane < 16) atomicAdd(s_red + ni * 16 + ra, s);  // cross-wave ds_add
  }
  __syncthreads();
  if (tid < NOUT)
    outp[(size_t)b * NOUT + tid] = s_red[tid] + (float)NATOM * b2[tid];
}

// ---------------------------------------------------------------------------
extern "C" void kernel_launch(void* const* d_in, const int* in_sizes, int n_in,
                              void* d_out, int out_size, void* d_ws,
                              size_t ws_size, hipStream_t stream) {
  (void)in_sizes; (void)n_in; (void)out_size; (void)d_ws; (void)ws_size;
  const float* atomf = (const float*)d_in[0];
  const float* bondf = (const float*)d_in[1];
  const float* adjf  = (const float*)d_in[2];
  const float* W1    = (const float*)d_in[3];
  const float* b1    = (const float*)d_in[4];
  const float* W2    = (const float*)d_in[5];
  const float* b2    = (const float*)d_in[6];
  float* outp        = (float*)d_out;

  ngf_fused<<<dim3(BATCH), dim3(256), LDS_BYTES, stream>>>(
      atomf, bondf, adjf, W1, b1, W2, b2, outp);
}